// GAT_65094524338334
// MI455X (gfx1250) — compile-verified
//
#include <hip/hip_runtime.h>
#include <hip/hip_bf16.h>

// ---- problem constants (match reference) ----
#define GAT_IN    128
#define GAT_H1    256      // HEADS*HID
#define GAT_HEADS 8
#define GAT_HID   32
#define GAT_OUT   8
#define NEG_SLOPE 0.2f

typedef float v2f __attribute__((ext_vector_type(2)));
typedef float v8f __attribute__((ext_vector_type(8)));

// ---------------------------------------------------------------------------
// float atomic max via signed-max / unsigned-min bit trick.
// Memory must be initialized to a negative-bit-pattern (we memset 0xFF).
// ---------------------------------------------------------------------------
__device__ __forceinline__ void atomicMaxF32(float* addr, float v) {
    if (v >= 0.0f) atomicMax((int*)addr, __float_as_int(v));
    else           atomicMin((unsigned int*)addr, __float_as_uint(v));
}

// ---------------------------------------------------------------------------
// GEMM1: H1[N,256] = X[N,128] @ W1[128,256]  (fp32 WMMA 16x16x4)
// one wave computes one 16x16 tile; 16 N-tiles per M-tile.
// ---------------------------------------------------------------------------
__global__ __launch_bounds__(256) void gat_gemm1_wmma(
        const float* __restrict__ X, const float* __restrict__ W,
        float* __restrict__ H, int nTiles) {
    const int wave = threadIdx.x >> 5;
    const int lane = threadIdx.x & 31;
    const int tile = blockIdx.x * 8 + wave;
    if (tile >= nTiles) return;                 // wave-uniform
    const int mt = tile >> 4;                   // M tile
    const int nt = tile & 15;                   // N tile
    const int m0 = mt * 16, n0 = nt * 16;
    const int lm = lane & 15;                   // row in A / col in B
    const int lk = (lane >> 4) * 2;             // K sub-pair: 0 or 2
    v8f acc = {};
    const float* __restrict__ Xr = X + (size_t)(m0 + lm) * GAT_IN;
    for (int k0 = 0; k0 < GAT_IN; k0 += 4) {
        v2f a, b;
        a.x = Xr[k0 + lk];
        a.y = Xr[k0 + lk + 1];
        b.x = W[(size_t)(k0 + lk)     * GAT_H1 + n0 + lm];
        b.y = W[(size_t)(k0 + lk + 1) * GAT_H1 + n0 + lm];
        acc = __builtin_amdgcn_wmma_f32_16x16x4_f32(
                false, a, false, b, (short)0, acc, false, false);
    }
    const int col   = n0 + lm;
    const int rbase = m0 + (lane >> 4) * 8;
#pragma unroll
    for (int i = 0; i < 8; ++i)
        H[(size_t)(rbase + i) * GAT_H1 + col] = acc[i];
}

// ---------------------------------------------------------------------------
// GEMM2: H2[N,8] = Hrelu[N,256] @ W2[256,8]  (fp32 WMMA, cols 8..15 padded 0)
// ---------------------------------------------------------------------------
__global__ __launch_bounds__(256) void gat_gemm2_wmma(
        const float* __restrict__ Hrelu, const float* __restrict__ W2,
        float* __restrict__ H2, int nMt) {
    const int wave = threadIdx.x >> 5;
    const int lane = threadIdx.x & 31;
    const int mt = blockIdx.x * 8 + wave;
    if (mt >= nMt) return;                      // wave-uniform
    const int m0 = mt * 16;
    const int lm = lane & 15;
    const int lk = (lane >> 4) * 2;
    v8f acc = {};
    const float* __restrict__ Xr = Hrelu + (size_t)(m0 + lm) * GAT_H1;
    const bool valid = (lm < GAT_OUT);
    for (int k0 = 0; k0 < GAT_H1; k0 += 4) {
        v2f a, b;
        a.x = Xr[k0 + lk];
        a.y = Xr[k0 + lk + 1];
        b.x = valid ? W2[(size_t)(k0 + lk)     * GAT_OUT + lm] : 0.0f;
        b.y = valid ? W2[(size_t)(k0 + lk + 1) * GAT_OUT + lm] : 0.0f;
        acc = __builtin_amdgcn_wmma_f32_16x16x4_f32(
                false, a, false, b, (short)0, acc, false, false);
    }
    if (valid) {
        const int rbase = m0 + (lane >> 4) * 8;
#pragma unroll
        for (int i = 0; i < 8; ++i)
            H2[(size_t)(rbase + i) * GAT_OUT + lm] = acc[i];
    }
}

// ---------------------------------------------------------------------------
// attention coefficients, layer 1: as/ad[n,h] = sum_c H1[n,h,c]*att[h,c]
// ---------------------------------------------------------------------------
__global__ void gat_att1(const float* __restrict__ H,
                         const float* __restrict__ att_s,
                         const float* __restrict__ att_d,
                         float* __restrict__ as_, float* __restrict__ ad_, int N) {
    int i = blockIdx.x * blockDim.x + threadIdx.x;
    if (i >= N * GAT_HEADS) return;
    const int n = i >> 3, h = i & 7;
    const float* hp = H + (size_t)n * GAT_H1 + h * GAT_HID;
    const float* sp = att_s + h * GAT_HID;
    const float* dp = att_d + h * GAT_HID;
    float accs = 0.f, accd = 0.f;
#pragma unroll
    for (int c = 0; c < GAT_HID; ++c) { float v = hp[c]; accs += v * sp[c]; accd += v * dp[c]; }
    as_[i] = accs; ad_[i] = accd;
}

// attention coefficients, layer 2 (1 head, 8 ch)
__global__ void gat_att2(const float* __restrict__ H2,
                         const float* __restrict__ att_s,
                         const float* __restrict__ att_d,
                         float* __restrict__ as_, float* __restrict__ ad_, int N) {
    int n = blockIdx.x * blockDim.x + threadIdx.x;
    if (n >= N) return;
    const float* hp = H2 + (size_t)n * GAT_OUT;
    float accs = 0.f, accd = 0.f;
#pragma unroll
    for (int c = 0; c < GAT_OUT; ++c) { float v = hp[c]; accs += v * att_s[c]; accd += v * att_d[c]; }
    as_[n] = accs; ad_[n] = accd;
}

// ---------------------------------------------------------------------------
// edge passes, layer 1 (8 heads)
// ---------------------------------------------------------------------------
__device__ __forceinline__ void edge_sd(const long long* __restrict__ ei,
                                        int e, int E, long long& s, long long& d) {
    if (e < E) { s = ei[e]; d = ei[(size_t)E + e]; }
    else       { s = d = e - E; }
}

__global__ void gat_edge_logit1(const long long* __restrict__ ei, int E, int N,
                                const float* __restrict__ as_, const float* __restrict__ ad_,
                                float* __restrict__ wbuf, float* __restrict__ emax) {
    int e = blockIdx.x * blockDim.x + threadIdx.x;
    if (e >= E + N) return;
    long long s, d; edge_sd(ei, e, E, s, d);
#pragma unroll
    for (int h = 0; h < GAT_HEADS; ++h) {
        float x = as_[s * GAT_HEADS + h] + ad_[d * GAT_HEADS + h];
        x = (x >= 0.f) ? x : NEG_SLOPE * x;
        wbuf[(size_t)e * GAT_HEADS + h] = x;
        atomicMaxF32(&emax[d * GAT_HEADS + h], x);
    }
}

__global__ void gat_edge_exp1(const long long* __restrict__ ei, int E, int N,
                              float* __restrict__ wbuf,
                              const float* __restrict__ emax, float* __restrict__ denom) {
    int e = blockIdx.x * blockDim.x + threadIdx.x;
    if (e >= E + N) return;
    long long s, d; edge_sd(ei, e, E, s, d); (void)s;
#pragma unroll
    for (int h = 0; h < GAT_HEADS; ++h) {
        float wv = expf(wbuf[(size_t)e * GAT_HEADS + h] - emax[d * GAT_HEADS + h]);
        wbuf[(size_t)e * GAT_HEADS + h] = wv;
        atomicAdd(&denom[d * GAT_HEADS + h], wv);
    }
}

// one block (256 threads) per edge: thread t handles channel t (head t>>5)
__global__ __launch_bounds__(256) void gat_edge_aggr1(
        const long long* __restrict__ ei, int E, int N,
        const float* __restrict__ wbuf, const float* __restrict__ denom,
        const float* __restrict__ H, float* __restrict__ agg) {
    int e = blockIdx.x;
    int t = threadIdx.x;
    long long s, d; edge_sd(ei, e, E, s, d);
    const int h = t >> 5;
    const float alpha = wbuf[(size_t)e * GAT_HEADS + h] / denom[d * GAT_HEADS + h];
    atomicAdd(&agg[(size_t)d * GAT_H1 + t], alpha * H[(size_t)s * GAT_H1 + t]);
}

__global__ void gat_relu_bias(const float* __restrict__ agg, const float* __restrict__ b,
                              float* __restrict__ out, int total) {
    int i = blockIdx.x * blockDim.x + threadIdx.x;
    if (i >= total) return;
    float v = agg[i] + b[i & (GAT_H1 - 1)];
    out[i] = (v > 0.f) ? v : 0.f;
}

// ---------------------------------------------------------------------------
// edge passes, layer 2 (1 head, 8 channels)
// ---------------------------------------------------------------------------
__global__ void gat_edge_logit2(const long long* __restrict__ ei, int E, int N,
                                const float* __restrict__ as_, const float* __restrict__ ad_,
                                float* __restrict__ wbuf, float* __restrict__ emax) {
    int e = blockIdx.x * blockDim.x + threadIdx.x;
    if (e >= E + N) return;
    long long s, d; edge_sd(ei, e, E, s, d);
    float x = as_[s] + ad_[d];
    x = (x >= 0.f) ? x : NEG_SLOPE * x;
    wbuf[e] = x;
    atomicMaxF32(&emax[d], x);
}

__global__ void gat_edge_exp2(const long long* __restrict__ ei, int E, int N,
                              float* __restrict__ wbuf,
                              const float* __restrict__ emax, float* __restrict__ denom) {
    int e = blockIdx.x * blockDim.x + threadIdx.x;
    if (e >= E + N) return;
    long long s, d; edge_sd(ei, e, E, s, d); (void)s;
    float wv = expf(wbuf[e] - emax[d]);
    wbuf[e] = wv;
    atomicAdd(&denom[d], wv);
}

__global__ void gat_edge_aggr2(const long long* __restrict__ ei, int E, int N,
                               const float* __restrict__ wbuf, const float* __restrict__ denom,
                               const float* __restrict__ H2, float* __restrict__ agg) {
    int e = blockIdx.x * blockDim.x + threadIdx.x;
    if (e >= E + N) return;
    long long s, d; edge_sd(ei, e, E, s, d);
    const float alpha = wbuf[e] / denom[d];
#pragma unroll
    for (int c = 0; c < GAT_OUT; ++c)
        atomicAdd(&agg[(size_t)d * GAT_OUT + c], alpha * H2[(size_t)s * GAT_OUT + c]);
}

__global__ void gat_bias2(const float* __restrict__ agg, const float* __restrict__ b,
                          float* __restrict__ out, int total) {
    int i = blockIdx.x * blockDim.x + threadIdx.x;
    if (i >= total) return;
    out[i] = agg[i] + b[i & (GAT_OUT - 1)];
}

// ---------------------------------------------------------------------------
extern "C" void kernel_launch(void* const* d_in, const int* in_sizes, int n_in,
                              void* d_out, int out_size, void* d_ws, size_t ws_size,
                              hipStream_t stream) {
    const float*     X   = (const float*)d_in[0];
    const long long* EI  = (const long long*)d_in[1];   // int64 [2,E]
    const float*     W1  = (const float*)d_in[2];
    const float*     As1 = (const float*)d_in[3];
    const float*     Ad1 = (const float*)d_in[4];
    const float*     B1  = (const float*)d_in[5];
    const float*     W2  = (const float*)d_in[6];
    const float*     As2 = (const float*)d_in[7];
    const float*     Ad2 = (const float*)d_in[8];
    const float*     B2  = (const float*)d_in[9];
    (void)n_in; (void)out_size; (void)ws_size;

    const int N    = in_sizes[0] / GAT_IN;     // 50000
    const int E    = in_sizes[1] / 2;          // 800000
    const int Etot = E + N;

    // ---- workspace carve-out ----
    char* w = (char*)d_ws;
    auto alloc = [&](size_t bytes) -> char* {
        char* p = w; w += (bytes + 255) & ~(size_t)255; return p;
    };
    float* H1     = (float*)alloc((size_t)N * GAT_H1 * 4);      // h, later relu(h)
    float* AGG1   = (float*)alloc((size_t)N * GAT_H1 * 4);
    float* as1    = (float*)alloc((size_t)N * GAT_HEADS * 4);
    float* ad1    = (float*)alloc((size_t)N * GAT_HEADS * 4);
    float* emax1  = (float*)alloc((size_t)N * GAT_HEADS * 4);
    float* denom1 = (float*)alloc((size_t)N * GAT_HEADS * 4);
    float* wbuf1  = (float*)alloc((size_t)Etot * GAT_HEADS * 4);
    float* H2     = (float*)alloc((size_t)N * GAT_OUT * 4);
    float* as2    = (float*)alloc((size_t)N * 4);
    float* ad2    = (float*)alloc((size_t)N * 4);
    float* emax2  = (float*)alloc((size_t)N * 4);
    float* denom2 = (float*)alloc((size_t)N * 4);
    float* wbuf2  = (float*)alloc((size_t)Etot * 4);
    float* AGG2   = (float*)alloc((size_t)N * GAT_OUT * 4);

    // ---- init accumulators (0xFF pattern == negative NaN bits: valid "-inf"
    //      sentinel for the bit-trick atomic max; overwritten via self-loops) ----
    hipMemsetAsync(emax1,  0xFF, (size_t)N * GAT_HEADS * 4, stream);
    hipMemsetAsync(denom1, 0,    (size_t)N * GAT_HEADS * 4, stream);
    hipMemsetAsync(AGG1,   0,    (size_t)N * GAT_H1 * 4,    stream);
    hipMemsetAsync(emax2,  0xFF, (size_t)N * 4,             stream);
    hipMemsetAsync(denom2, 0,    (size_t)N * 4,             stream);
    hipMemsetAsync(AGG2,   0,    (size_t)N * GAT_OUT * 4,   stream);

    const int TB = 256;
    const int edgeBlocks = (Etot + TB - 1) / TB;

    // ---- layer 1 ----
    {
        const int tiles = (N / 16) * (GAT_H1 / 16);           // 3125*16
        gat_gemm1_wmma<<<(tiles + 7) / 8, 256, 0, stream>>>(X, W1, H1, tiles);
    }
    gat_att1<<<(N * GAT_HEADS + TB - 1) / TB, TB, 0, stream>>>(H1, As1, Ad1, as1, ad1, N);
    gat_edge_logit1<<<edgeBlocks, TB, 0, stream>>>(EI, E, N, as1, ad1, wbuf1, emax1);
    gat_edge_exp1  <<<edgeBlocks, TB, 0, stream>>>(EI, E, N, wbuf1, emax1, denom1);
    gat_edge_aggr1 <<<Etot, 256, 0, stream>>>(EI, E, N, wbuf1, denom1, H1, AGG1);
    gat_relu_bias  <<<((size_t)N * GAT_H1 + TB - 1) / TB, TB, 0, stream>>>(AGG1, B1, H1, N * GAT_H1);

    // ---- layer 2 ----
    {
        const int nMt = N / 16;                               // 3125
        gat_gemm2_wmma<<<(nMt + 7) / 8, 256, 0, stream>>>(H1, W2, H2, nMt);
    }
    gat_att2<<<(N + TB - 1) / TB, TB, 0, stream>>>(H2, As2, Ad2, as2, ad2, N);
    gat_edge_logit2<<<edgeBlocks, TB, 0, stream>>>(EI, E, N, as2, ad2, wbuf2, emax2);
    gat_edge_exp2  <<<edgeBlocks, TB, 0, stream>>>(EI, E, N, wbuf2, emax2, denom2);
    gat_edge_aggr2 <<<edgeBlocks, TB, 0, stream>>>(EI, E, N, wbuf2, denom2, H2, AGG2);
    gat_bias2      <<<(N * GAT_OUT + TB - 1) / TB, TB, 0, stream>>>(AGG2, B2, (float*)d_out, N * GAT_OUT);
}